// SparseMoE_5506148073585
// MI455X (gfx1250) — compile-verified
//
#include <hip/hip_runtime.h>
#include <hip/hip_bf16.h>
#include <math.h>

typedef __attribute__((ext_vector_type(2))) float v2f;
typedef __attribute__((ext_vector_type(8))) float v8f;

#define T_TOK 8192
#define DDIM  1024
#define HDIM  4096
#define NEXP  8
#define MT    64           // tokens (M) per tile: B-fragments reused across 4 M-subtiles
#define HC    128          // H chunk per pass (8 waves x 16 cols)
#define XLD   1032         // padded LDS row stride for X tile (8-bank skew per row)
#define HLD   136          // padded LDS row stride for H tile
#define FFN_SMEM_BYTES ((MT * XLD + MT * HLD) * 4 + MT * 8)

__device__ __forceinline__ v8f wmma_f32(v2f a, v2f b, v8f c) {
    // D = A(16x4 f32) * B(4x16 f32) + C(16x16 f32)
    return __builtin_amdgcn_wmma_f32_16x16x4_f32(
        /*neg_a=*/false, a, /*neg_b=*/false, b,
        /*c_mod=*/(short)0, c, /*reuse_a=*/false, /*reuse_b=*/false);
}

// ---------------------------------------------------------------------------
// Kernel 0: zero the output (scatter-add target) and expert counters
// ---------------------------------------------------------------------------
__global__ void moe_zero_kernel(float* __restrict__ out, size_t n, int* __restrict__ counts) {
    size_t i = (size_t)blockIdx.x * blockDim.x + threadIdx.x;
    size_t stride = (size_t)gridDim.x * blockDim.x;
    for (; i < n; i += stride) out[i] = 0.0f;
    if (blockIdx.x == 0 && threadIdx.x < NEXP) counts[threadIdx.x] = 0;
}

// ---------------------------------------------------------------------------
// Kernel 1: noisy top-2 router; builds per-expert (token, gate) lists
// One 256-thread block per token.
// ---------------------------------------------------------------------------
__global__ void moe_router_kernel(const float* __restrict__ x,
                                  const float* __restrict__ eps,
                                  const float* __restrict__ Wg, const float* __restrict__ bg,
                                  const float* __restrict__ Wn, const float* __restrict__ bn,
                                  int* __restrict__ counts,
                                  int* __restrict__ tokenList,
                                  float* __restrict__ gateList) {
    __shared__ float red[8][16];
    __shared__ float fin[16];

    const int t   = blockIdx.x;
    const int tid = threadIdx.x;

    float accG[NEXP] = {0.f,0.f,0.f,0.f,0.f,0.f,0.f,0.f};
    float accN[NEXP] = {0.f,0.f,0.f,0.f,0.f,0.f,0.f,0.f};
    const float* xt = x + (size_t)t * DDIM;

    for (int d = tid; d < DDIM; d += 256) {
        const float xv = xt[d];
        const float* wg = Wg + (size_t)d * NEXP;  // Wg [D,E] row-major
        const float* wn = Wn + (size_t)d * NEXP;
        #pragma unroll
        for (int e = 0; e < NEXP; ++e) {
            accG[e] = fmaf(xv, wg[e], accG[e]);
            accN[e] = fmaf(xv, wn[e], accN[e]);
        }
    }
    // intra-wave reduce (wave32)
    #pragma unroll
    for (int e = 0; e < NEXP; ++e) {
        #pragma unroll
        for (int off = 16; off > 0; off >>= 1) {
            accG[e] += __shfl_xor(accG[e], off, 32);
            accN[e] += __shfl_xor(accN[e], off, 32);
        }
    }
    const int wave = tid >> 5, lane = tid & 31;
    if (lane == 0) {
        #pragma unroll
        for (int e = 0; e < NEXP; ++e) { red[wave][e] = accG[e]; red[wave][NEXP + e] = accN[e]; }
    }
    __syncthreads();
    if (tid < 16) {
        float s = 0.f;
        #pragma unroll
        for (int w = 0; w < 8; ++w) s += red[w][tid];
        fin[tid] = s;
    }
    __syncthreads();
    if (tid == 0) {
        float noisy[NEXP];
        const float* ept = eps + (size_t)t * NEXP;
        #pragma unroll
        for (int e = 0; e < NEXP; ++e) {
            const float g  = fin[e] + bg[e];
            const float nl = fin[NEXP + e] + bn[e];
            // stable softplus
            const float sp = fmaxf(nl, 0.0f) + log1pf(expf(-fabsf(nl)));
            noisy[e] = g + ept[e] * sp;
        }
        int i0 = 0;
        #pragma unroll
        for (int e = 1; e < NEXP; ++e) if (noisy[e] > noisy[i0]) i0 = e;
        int i1 = (i0 == 0) ? 1 : 0;
        #pragma unroll
        for (int e = 0; e < NEXP; ++e) if (e != i0 && noisy[e] > noisy[i1]) i1 = e;
        const float ex = expf(noisy[i1] - noisy[i0]);   // <= 1
        const float g0 = 1.0f / (1.0f + ex);
        const float g1 = ex / (1.0f + ex);
        int p0 = atomicAdd(&counts[i0], 1);
        tokenList[i0 * T_TOK + p0] = t;  gateList[i0 * T_TOK + p0] = g0;
        int p1 = atomicAdd(&counts[i1], 1);
        tokenList[i1 * T_TOK + p1] = t;  gateList[i1 * T_TOK + p1] = g1;
    }
}

// ---------------------------------------------------------------------------
// Kernel 2: per-expert FFN via FP32 WMMA; persistent blocks over 64-token tiles.
// B-fragments (expert weights) are loaded once per wave and reused across the
// 4 M-subtiles in registers -> 4x less L2 weight traffic than M=16 tiling.
// ---------------------------------------------------------------------------
__launch_bounds__(256, 1)
__global__ void moe_ffn_kernel(const float* __restrict__ x,
                               const float* __restrict__ W1, const float* __restrict__ b1,
                               const float* __restrict__ W2, const float* __restrict__ b2,
                               const int* __restrict__ counts,
                               const int* __restrict__ tokenList,
                               const float* __restrict__ gateList,
                               float* __restrict__ out) {
    extern __shared__ float smem[];
    float* Xs     = smem;                          // MT x XLD   (~258 KB)
    float* Hs     = smem + MT * XLD;               // MT x HLD   (~34 KB)
    int*   tokenS = (int*)(Hs + MT * HLD);         // MT
    float* gateS  = (float*)(tokenS + MT);         // MT

    const int tid   = threadIdx.x;
    const int wave  = tid >> 5;
    const int lane  = tid & 31;
    const int lhalf = lane >> 4;   // 0|1: K-pair select / M upper half (ISA layout)
    const int lmod  = lane & 15;   // M (A) or N (B,C)

    int cnt[NEXP], pre[NEXP + 1];
    pre[0] = 0;
    #pragma unroll
    for (int e = 0; e < NEXP; ++e) {
        cnt[e] = counts[e];
        pre[e + 1] = pre[e] + ((cnt[e] + MT - 1) >> 6);
    }
    const int totalTiles = pre[NEXP];

    for (int tile = blockIdx.x; tile < totalTiles; tile += gridDim.x) {
        int e = 0;
        while (tile >= pre[e + 1]) ++e;
        const int base = (tile - pre[e]) * MT;
        const int rows = min(MT, cnt[e] - base);

        if (tid < MT) {
            const int safe = (tid < rows) ? tid : 0;   // pad rows reuse row 0, gate 0
            tokenS[tid] = tokenList[e * T_TOK + base + safe];
            gateS[tid]  = (tid < rows) ? gateList[e * T_TOK + base + tid] : 0.0f;
        }
        __syncthreads();

        // gather X tile (float4 vectorized): MT rows x 256 float4
        for (int i = tid; i < MT * (DDIM / 4); i += 256) {
            const int m  = i >> 8;            // DDIM/4 == 256
            const int c4 = i & 255;
            const float4 v = ((const float4*)(x + (size_t)tokenS[m] * DDIM))[c4];
            *(float4*)&Xs[m * XLD + c4 * 4] = v;
        }
        __syncthreads();

        const float* W1e = W1 + (size_t)e * DDIM * HDIM;   // [D,H]
        const float* W2e = W2 + (size_t)e * HDIM * DDIM;   // [H,D]
        const float* b1e = b1 + (size_t)e * HDIM;
        const float* b2e = b2 + (size_t)e * DDIM;

        v8f acc[4][8];   // O[64,1024]: 4 M-subtiles x 8 N-tiles per wave (256 VGPRs)
        #pragma unroll
        for (int ms = 0; ms < 4; ++ms)
            #pragma unroll
            for (int i = 0; i < 8; ++i)
                acc[ms][i] = (v8f){0.f,0.f,0.f,0.f,0.f,0.f,0.f,0.f};

        for (int hc = 0; hc < HDIM; hc += HC) {
            // ---- GEMM1: H[64,HC] = X[64,1024] @ W1[:,hc:hc+HC]
            //      wave w owns one 16-wide column group; B loaded once per k-step,
            //      reused over 4 M-subtiles.
            v8f hacc[4];
            #pragma unroll
            for (int ms = 0; ms < 4; ++ms)
                hacc[ms] = (v8f){0.f,0.f,0.f,0.f,0.f,0.f,0.f,0.f};
            const int ncol = hc + wave * 16 + lmod;
            #pragma unroll 2
            for (int k = 0; k < DDIM; k += 4) {
                const int ka = k + 2 * lhalf;
                v2f b;
                b.x = W1e[(size_t)(ka    ) * HDIM + ncol];   // B: K=ka,   N=ncol
                b.y = W1e[(size_t)(ka + 1) * HDIM + ncol];   // B: K=ka+1, N=ncol
                #pragma unroll
                for (int ms = 0; ms < 4; ++ms) {
                    v2f a = *(const v2f*)&Xs[(ms * 16 + lmod) * XLD + ka];
                    hacc[ms] = wmma_f32(a, b, hacc[ms]);
                }
            }
            const float bias = b1e[ncol];
            #pragma unroll
            for (int ms = 0; ms < 4; ++ms) {
                #pragma unroll
                for (int r = 0; r < 8; ++r) {
                    float v = hacc[ms][r] + bias;            // row M = ms*16 + r + 8*lhalf
                    v = 0.5f * v * (1.0f + erff(v * 0.70710678118654752f)); // exact GELU
                    Hs[(ms * 16 + r + 8 * lhalf) * HLD + wave * 16 + lmod] = v;
                }
            }
            __syncthreads();

            // ---- GEMM2: O[64,1024] += H[64,HC] @ W2[hc:hc+HC,:]
            //      wave w owns 8 N-tiles; each B-fragment reused over 4 M-subtiles.
            for (int k = 0; k < HC; k += 4) {
                const int ka = k + 2 * lhalf;
                v2f a[4];
                #pragma unroll
                for (int ms = 0; ms < 4; ++ms)
                    a[ms] = *(const v2f*)&Hs[(ms * 16 + lmod) * HLD + ka];
                #pragma unroll
                for (int i = 0; i < 8; ++i) {
                    const int nb = (wave * 8 + i) * 16 + lmod;
                    v2f b;
                    b.x = W2e[(size_t)(hc + ka    ) * DDIM + nb];
                    b.y = W2e[(size_t)(hc + ka + 1) * DDIM + nb];
                    #pragma unroll
                    for (int ms = 0; ms < 4; ++ms)
                        acc[ms][i] = wmma_f32(a[ms], b, acc[ms][i]);
                }
            }
            __syncthreads();
        }

        // ---- (acc + b2) * gate, scatter-add to out
        #pragma unroll
        for (int i = 0; i < 8; ++i) {
            const int nb = (wave * 8 + i) * 16 + lmod;
            const float b2v = b2e[nb];
            #pragma unroll
            for (int ms = 0; ms < 4; ++ms) {
                #pragma unroll
                for (int r = 0; r < 8; ++r) {
                    const int m = ms * 16 + r + 8 * lhalf;
                    if (m < rows) {
                        const float val = (acc[ms][i][r] + b2v) * gateS[m];
                        atomicAdd(&out[(size_t)tokenS[m] * DDIM + nb], val);
                    }
                }
            }
        }
        __syncthreads();   // protect tokenS/gateS/Xs/Hs for next tile
    }
}

// ---------------------------------------------------------------------------
extern "C" void kernel_launch(void* const* d_in, const int* in_sizes, int n_in,
                              void* d_out, int out_size, void* d_ws, size_t ws_size,
                              hipStream_t stream) {
    const float* x   = (const float*)d_in[0];
    const float* eps = (const float*)d_in[1];
    const float* Wg  = (const float*)d_in[2];
    const float* bg  = (const float*)d_in[3];
    const float* Wn  = (const float*)d_in[4];
    const float* bn  = (const float*)d_in[5];
    const float* W1  = (const float*)d_in[6];
    const float* b1  = (const float*)d_in[7];
    const float* W2  = (const float*)d_in[8];
    const float* b2  = (const float*)d_in[9];
    // d_in[10] = top_k (== 2, compiled in)
    float* out = (float*)d_out;

    // workspace layout
    int*   counts    = (int*)d_ws;                                      // 8 ints (pad 256B)
    int*   tokenList = (int*)((char*)d_ws + 256);                       // E*T ints
    float* gateList  = (float*)((char*)d_ws + 256 + NEXP * T_TOK * 4);  // E*T floats

    // opt in to >64KB dynamic LDS (320KB/WGP on gfx1250); deterministic no-op if redundant
    (void)hipFuncSetAttribute((const void*)moe_ffn_kernel,
                              hipFuncAttributeMaxDynamicSharedMemorySize,
                              FFN_SMEM_BYTES);

    moe_zero_kernel<<<2048, 256, 0, stream>>>(out, (size_t)T_TOK * DDIM, counts);
    moe_router_kernel<<<T_TOK, 256, 0, stream>>>(x, eps, Wg, bg, Wn, bn,
                                                 counts, tokenList, gateList);
    // max tiles = ceil(2T/64) + (E-1) slack = 263; 272 persistent blocks
    moe_ffn_kernel<<<272, 256, FFN_SMEM_BYTES, stream>>>(x, W1, b1, W2, b2,
                                                         counts, tokenList, gateList, out);
}